// LIF_70824010711791
// MI455X (gfx1250) — compile-verified
//
#include <hip/hip_runtime.h>

// LIF forward scan, T=8, decay=0.25, threshold=0.5.
// Pure streaming kernel: 268 MB in + 268 MB out, zero reuse -> HBM-bound
// (~23 us floor at 23.3 TB/s). Strategy: 4 consecutive f32 per thread, all 8
// timestep loads issued up front (independent of scan state) as non-temporal
// b128 loads; scan runs in registers; spikes stored as non-temporal b128.

#define LIF_T 8
#define LIF_THRESH 0.5f
#define LIF_DECAY 0.25f

// Native clang vector type: required by __builtin_nontemporal_load/store
// (HIP's float4 is a class and is rejected by the builtin).
typedef float v4f __attribute__((ext_vector_type(4)));

__global__ __launch_bounds__(256) void lif_scan_kernel(
    const v4f* __restrict__ x,    // [T * n4] float4-vectors
    v4f* __restrict__ spikes,     // [T * n4] float4-vectors
    int n4)                       // elements-per-timestep / 4
{
    const int i = blockIdx.x * blockDim.x + threadIdx.x;
    if (i >= n4) return;

    // Stage all T timesteps for this element group: 8 independent
    // global_load_b128 (TH=NT) in flight -> latency hidden by MLP, and the
    // streaming data does not displace L2 (footprint >> 192 MB anyway).
    v4f xt[LIF_T];
#pragma unroll
    for (int t = 0; t < LIF_T; ++t) {
        xt[t] = __builtin_nontemporal_load(x + (size_t)t * (size_t)n4 + i);
    }

    // LIF scan in registers. Note v*(1-s) with s in {0,1} == select(v>=thr, 0, v)
    // for finite v (random-normal inputs), lowering to v_cmp + v_cndmask.
    float vx = 0.0f, vy = 0.0f, vz = 0.0f, vw = 0.0f;
#pragma unroll
    for (int t = 0; t < LIF_T; ++t) {
        vx = __builtin_fmaf(vx, LIF_DECAY, xt[t].x);
        vy = __builtin_fmaf(vy, LIF_DECAY, xt[t].y);
        vz = __builtin_fmaf(vz, LIF_DECAY, xt[t].z);
        vw = __builtin_fmaf(vw, LIF_DECAY, xt[t].w);

        const bool fx = (vx >= LIF_THRESH);
        const bool fy = (vy >= LIF_THRESH);
        const bool fz = (vz >= LIF_THRESH);
        const bool fw = (vw >= LIF_THRESH);

        v4f s4;
        s4.x = fx ? 1.0f : 0.0f;
        s4.y = fy ? 1.0f : 0.0f;
        s4.z = fz ? 1.0f : 0.0f;
        s4.w = fw ? 1.0f : 0.0f;

        vx = fx ? 0.0f : vx;
        vy = fy ? 0.0f : vy;
        vz = fz ? 0.0f : vz;
        vw = fw ? 0.0f : vw;

        __builtin_nontemporal_store(s4, spikes + (size_t)t * (size_t)n4 + i);
    }
}

extern "C" void kernel_launch(void* const* d_in, const int* in_sizes, int n_in,
                              void* d_out, int out_size, void* d_ws, size_t ws_size,
                              hipStream_t stream) {
    (void)n_in; (void)out_size; (void)d_ws; (void)ws_size;

    const float* x = (const float*)d_in[0];
    float* out = (float*)d_out;

    // in_sizes[0] = T*B*C*H*W total elements; per-timestep element count is
    // divisible by 4 (H*W = 1024), so 16-byte vector addressing is exact.
    const int total = in_sizes[0];
    const int per_t = total / LIF_T;
    const int n4 = per_t / 4;

    const int block = 256;
    const int grid = (n4 + block - 1) / block;

    lif_scan_kernel<<<grid, block, 0, stream>>>(
        (const v4f*)x, (v4f*)out, n4);
}